// SKA_PyTorch_1013612282168
// MI455X (gfx1250) — compile-verified
//
#include <hip/hip_runtime.h>
#include <hip/hip_bf16.h>
#include <stdint.h>

// SKA forward (per-pixel 3x3 stencil, weights shared across 32 groups).
// Memory-bound: ~234 MB total traffic => ~10us floor at 23.3 TB/s.
// CDNA5 path: TDM tensor_load_to_lds stages each contiguous 12.5KB x-plane
// into LDS once; all 9 stencil taps are then served from LDS.

typedef uint32_t v4u __attribute__((ext_vector_type(4)));
typedef int32_t  v8i __attribute__((ext_vector_type(8)));
typedef int32_t  v4i __attribute__((ext_vector_type(4)));
typedef float    f4  __attribute__((ext_vector_type(4)));

#define SKA_B      16
#define SKA_C      512
#define SKA_CG     16
#define SKA_H      56
#define SKA_W      56
#define SKA_HW     (SKA_H * SKA_W)       // 3136 elements = 12544 bytes per plane
#define SKA_HW4    (SKA_HW / 4)          // 784 float4s per plane (56 % 4 == 0)
#define SKA_W4     (SKA_W / 4)           // 14 float4s per row
#define SKA_K2     9
#define SKA_NBLK   (SKA_B * SKA_C)       // 8192 blocks: one (batch, channel) plane each
#define SKA_THREADS 256

__global__ __launch_bounds__(SKA_THREADS)
void ska_fwd_kernel(const float* __restrict__ x,
                    const float* __restrict__ w,
                    float* __restrict__ out)
{
    __shared__ float xs[SKA_HW];         // 12544 B of LDS -> high occupancy (320KB WGP)

    const int bci = blockIdx.x;          // b*512 + channel
    const int b   = bci >> 9;            // / 512
    const int ci  = bci & 511;
    const int c   = ci & (SKA_CG - 1);   // channel-in-group (weights shared over groups)

    const float* __restrict__ xplane = x + (size_t)bci * SKA_HW;
    const f4*    __restrict__ wbase4 = (const f4*)(w + ((size_t)(b * SKA_CG + c) * SKA_K2) * SKA_HW);
    f4*          __restrict__ oplane4 = (f4*)(out + (size_t)bci * SKA_HW);

    const int tid = threadIdx.x;

#if __has_builtin(__builtin_amdgcn_tensor_load_to_lds) && __has_builtin(__builtin_amdgcn_s_wait_tensorcnt)
    // Wave 0 issues one TDM descriptor copying the whole contiguous plane.
    if (tid < 32) {
        const uint32_t lds_off = (uint32_t)(uintptr_t)(&xs[0]);       // flat->LDS offset (low 32 bits)
        const uint64_t gaddr   = (uint64_t)(uintptr_t)xplane;

        v4u g0;
        g0.x = 1u;                                                    // count=1, user descriptor, no gather
        g0.y = lds_off;                                               // lds_addr
        g0.z = (uint32_t)(gaddr & 0xFFFFFFFFu);                       // global_addr[31:0]
        g0.w = (uint32_t)((gaddr >> 32) & 0x1FFFFFFu) | (2u << 30);   // global_addr[56:32] | type=2 ("image")

        v8i g1;
        g1[0] = (int)(2u << 16);                    // workgroup_mask=0, data_size=2 (4 bytes), no pad/iterate
        g1[1] = (int)((uint32_t)SKA_HW << 16);      // tensor_dim0[15:0] (bits [79:48] of group1)
        g1[2] = (int)(1u << 16);                    // tensor_dim0 hi = 0 ; tensor_dim1 = 1
        g1[3] = (int)((uint32_t)SKA_HW << 16);      // tensor_dim1 hi = 0 ; tile_dim0 = 3136
        g1[4] = 0;                                  // tile_dim1 = 0, tile_dim2 = 0 (1-D tile)
        g1[5] = SKA_HW;                             // tensor_dim0_stride[31:0]
        g1[6] = (int)((uint32_t)SKA_HW << 16);      // dim0_stride hi = 0 ; tensor_dim1_stride[15:0]
        g1[7] = 0;
        v4i gz4 = {0, 0, 0, 0};                     // groups 2/3 unused (<=2D tensor)
        v8i gz8 = {0, 0, 0, 0, 0, 0, 0, 0};         // extra group (6-arg builtin form), unused

        __builtin_amdgcn_tensor_load_to_lds(g0, g1, gz4, gz4, gz8, 0);
        __builtin_amdgcn_s_wait_tensorcnt(0);       // plane resident in LDS
    }
#else
    for (int p = tid; p < SKA_HW; p += SKA_THREADS) xs[p] = xplane[p];
#endif
    __syncthreads();

    // Each thread computes ~3 float4 output packs (784 = 3*256 + 16).
    // A float4 never crosses a row (56 % 4 == 0), so all 4 pixels share h.
    for (int q = tid; q < SKA_HW4; q += SKA_THREADS) {
        const int h    = q / SKA_W4;                 // row
        const int col0 = (q - h * SKA_W4) * 4;       // first column of the pack

        // Per-tap per-pixel weights: 9 coalesced b128 loads (kept L2-resident,
        // reused by the 32 group-sharing blocks).
        f4 wv[SKA_K2];
#pragma unroll
        for (int k = 0; k < SKA_K2; ++k)
            wv[k] = wbase4[(size_t)k * SKA_HW4 + q];

        // 3 rows x 6 columns of x halo from LDS, zero-padded at the borders.
        float xr[3][6];
#pragma unroll
        for (int r = 0; r < 3; ++r) {
            const int  hh    = h - 1 + r;
            const bool rowok = (hh >= 0) & (hh < SKA_H);
            const int  base  = hh * SKA_W + col0;
            xr[r][0] = (rowok && col0 > 0) ? xs[base - 1] : 0.0f;
#pragma unroll
            for (int j = 0; j < 4; ++j)
                xr[r][1 + j] = rowok ? xs[base + j] : 0.0f;
            xr[r][5] = (rowok && (col0 + 4) < SKA_W) ? xs[base + 4] : 0.0f;
        }

        // out[j] = sum_{r,d} w[r*3+d][j] * xr[r][j+d]
        f4 acc = {0.0f, 0.0f, 0.0f, 0.0f};
#pragma unroll
        for (int r = 0; r < 3; ++r) {
#pragma unroll
            for (int d = 0; d < 3; ++d) {
                const f4 wk = wv[r * 3 + d];
                acc[0] = fmaf(xr[r][0 + d], wk[0], acc[0]);
                acc[1] = fmaf(xr[r][1 + d], wk[1], acc[1]);
                acc[2] = fmaf(xr[r][2 + d], wk[2], acc[2]);
                acc[3] = fmaf(xr[r][3 + d], wk[3], acc[3]);
            }
        }

        // Stream output past L2 (NT) so the reused w planes stay resident.
        __builtin_nontemporal_store(acc, &oplane4[q]);
    }
}

extern "C" void kernel_launch(void* const* d_in, const int* in_sizes, int n_in,
                              void* d_out, int out_size, void* d_ws, size_t ws_size,
                              hipStream_t stream) {
    const float* x = (const float*)d_in[0];   // (16, 512, 56, 56) fp32
    const float* w = (const float*)d_in[1];   // (16, 16, 9, 56, 56) fp32
    float* out = (float*)d_out;               // (16, 512, 56, 56) fp32

    ska_fwd_kernel<<<dim3(SKA_NBLK), dim3(SKA_THREADS), 0, stream>>>(x, w, out);
}